// TwoDCausalSelfAttention_84507776516554
// MI455X (gfx1250) — compile-verified
//
#include <hip/hip_runtime.h>
#include <hip/hip_bf16.h>

// ---------------------------------------------------------------------------
// TwoDCausalSelfAttention for MI455X (gfx1250, wave32, WMMA)
//   B=2, T=1024, C=256, head_dim=2, H=128
// Pipeline:
//   1) qkv = x @ w_qkv^T          (WMMA f32 16x16x4, 16x32 tile per wave)
//   2) RMSNorm(q,k) per head pair (elementwise, d=2)
//   3) flash attention per (b,h)  (K/V async-staged in LDS, 2-pass softmax)
//   4) out = y @ w_proj^T         (WMMA f32 16x16x4)
// ---------------------------------------------------------------------------

typedef __attribute__((ext_vector_type(2))) float v2f;
typedef __attribute__((ext_vector_type(8))) float v8f;
typedef int v2i __attribute__((vector_size(8)));   // matches builtin param type

#define BB 2
#define TT 1024
#define CC 256
#define HH 128      // n_head
#define QKV_N (3*CC)

// Async global->LDS staging (ASYNCcnt path), used only if the toolchain has
// both the copy and the wait builtins; otherwise fall back to VGPR round-trip.
#if defined(__HIP_DEVICE_COMPILE__) && \
    __has_builtin(__builtin_amdgcn_global_load_async_to_lds_b64) && \
    __has_builtin(__builtin_amdgcn_s_wait_asynccnt)
#define USE_ASYNC_LDS 1
#else
#define USE_ASYNC_LDS 0
#endif

#define AS1 __attribute__((address_space(1)))
#define AS3 __attribute__((address_space(3)))

// ---------------------------------------------------------------------------
// GEMM: C(M,N) = A(M,K) @ W(N,K)^T, all f32 row-major.
// One wave32 computes a 16(M) x 32(N) tile of C via two chained
// V_WMMA_F32_16X16X4_F32 per K-step, sharing one A fragment.
//
// A-fragment layout (16x4 f32, ISA 7.12.2): lanes 0-15 -> rows M=0..15 with
// {K=0,K=1} in the two VGPRs; lanes 16-31 -> same rows with {K=2,K=3}.
// B-fragment (4x16) is symmetric with N across lanes. Both map to a single
// contiguous float2 load per lane per K-step.
// C/D layout: lane<16: N=lane, VGPR r -> M=r; lane>=16: N=lane-16, M=r+8.
// ---------------------------------------------------------------------------
__global__ __launch_bounds__(256) void wmma_gemm_xwt(
    const float* __restrict__ A,   // (M,K)
    const float* __restrict__ W,   // (N,K)
    float* __restrict__ C,         // (M,N)
    int M, int N, int K)
{
    const int wave   = (blockIdx.x * blockDim.x + threadIdx.x) >> 5;
    const int lane   = threadIdx.x & 31;
    const int ntile2 = N >> 5;                 // 32-wide N tiles
    const int mt = (wave / ntile2) << 4;
    const int nt = (wave % ntile2) << 5;
    if (mt >= M) return;

    const int hl = lane >> 4;      // 0: K pair {0,1}, 1: K pair {2,3}
    const int l  = lane & 15;

    const float* arow  = A + (size_t)(mt + l) * K + 2 * hl;
    const float* b0row = W + (size_t)(nt + l) * K + 2 * hl;
    const float* b1row = W + (size_t)(nt + 16 + l) * K + 2 * hl;

    v8f acc0 = {};
    v8f acc1 = {};
#pragma unroll 8
    for (int k = 0; k < K; k += 4) {
        v2f a  = *(const v2f*)(arow + k);
        v2f b0 = *(const v2f*)(b0row + k);
        v2f b1 = *(const v2f*)(b1row + k);
        // 8 args: (neg_a, A, neg_b, B, c_mod, C, reuse_a, reuse_b)
        acc0 = __builtin_amdgcn_wmma_f32_16x16x4_f32(
            false, a, false, b0, (short)0, acc0, false, false);
        acc1 = __builtin_amdgcn_wmma_f32_16x16x4_f32(
            false, a, false, b1, (short)0, acc1, false, false);
    }

    float* c0 = C + (size_t)(mt + hl * 8) * N + nt + l;
#pragma unroll
    for (int r = 0; r < 8; ++r) {
        c0[(size_t)r * N]      = acc0[r];
        c0[(size_t)r * N + 16] = acc1[r];
    }
}

// ---------------------------------------------------------------------------
// RMSNorm over each (q,k) head pair (head_dim = 2), in-place on the qkv
// buffer. qkv layout: (B*T, 3C); q cols [0,256), k cols [256,512).
// y = x * rsqrt(mean(x^2) + eps) * w
// ---------------------------------------------------------------------------
__global__ __launch_bounds__(256) void rmsnorm_qk(
    float* __restrict__ qkv,
    const float* __restrict__ qn_w,
    const float* __restrict__ kn_w)
{
    const int i  = blockIdx.x * blockDim.x + threadIdx.x;   // 0 .. B*T*H-1
    const int bt = i >> 7;            // / HH
    const int h  = i & (HH - 1);
    const float eps = 1e-6f;

    float* q = qkv + (size_t)bt * QKV_N + 2 * h;
    float q0 = q[0], q1 = q[1];
    float qs = __frsqrt_rn(0.5f * (q0 * q0 + q1 * q1) + eps);
    q[0] = q0 * qs * qn_w[0];
    q[1] = q1 * qs * qn_w[1];

    float* k = qkv + (size_t)bt * QKV_N + CC + 2 * h;
    float k0 = k[0], k1 = k[1];
    float ks = __frsqrt_rn(0.5f * (k0 * k0 + k1 * k1) + eps);
    k[0] = k0 * ks * kn_w[0];
    k[1] = k1 * ks * kn_w[1];
}

// ---------------------------------------------------------------------------
// Causal flash attention, head_dim = 2. One workgroup (256 threads) per
// (b,h). K and V for the head (1024 x 2 each, 16 KB total) are staged in LDS
// (async global->LDS when available); each thread owns 4 query rows.
// Two-pass softmax per query: (1) unrolled fmax scan for the row max,
// (2) unrolled exp+FMA accumulation with independent partial sums. This
// breaks the long serial online-softmax recurrence while keeping exact
// reference semantics (up to FP sum reassociation).
// ---------------------------------------------------------------------------
__global__ __launch_bounds__(256) void attn_head(
    const float* __restrict__ qkv,   // (B*T, 3C), q/k already RMSNorm'd
    float* __restrict__ y)           // (B*T, C)
{
    __shared__ float kbuf[TT * 2];
    __shared__ float vbuf[TT * 2];

    const int bh = blockIdx.x;          // 0 .. B*H-1
    const int b  = bh >> 7;
    const int h  = bh & (HH - 1);
    const int tid = threadIdx.x;

    const float* base = qkv + (size_t)b * TT * QKV_N;

    // Stage K/V for this head into LDS (8 B per row per buffer).
    for (int t = tid; t < TT; t += 256) {
        const float* kp = base + (size_t)t * QKV_N + CC   + 2 * h;
        const float* vp = base + (size_t)t * QKV_N + 2*CC + 2 * h;
#if USE_ASYNC_LDS
        __builtin_amdgcn_global_load_async_to_lds_b64(
            (AS1 v2i*)kp, (AS3 v2i*)&kbuf[2 * t], 0, 0);
        __builtin_amdgcn_global_load_async_to_lds_b64(
            (AS1 v2i*)vp, (AS3 v2i*)&vbuf[2 * t], 0, 0);
#else
        kbuf[2 * t]     = kp[0];
        kbuf[2 * t + 1] = kp[1];
        vbuf[2 * t]     = vp[0];
        vbuf[2 * t + 1] = vp[1];
#endif
    }
#if USE_ASYNC_LDS
    __builtin_amdgcn_s_wait_asynccnt(0);
#endif
    __syncthreads();

    const v2f* kk = (const v2f*)kbuf;
    const v2f* vv = (const v2f*)vbuf;
    const float scale = 0.70710678118654752440f;   // 1/sqrt(head_dim)
    const float NEG_INF = -3.402823466e38f;

    for (int t = tid; t < TT; t += 256) {
        const float* qp = base + (size_t)t * QKV_N + 2 * h;
        const float q0 = qp[0] * scale;
        const float q1 = qp[1] * scale;
        const int n = t + 1;                    // causal: keys s in [0, t]

        // ---- pass 1: row max (independent fmax chains) ----
        float m0 = NEG_INF, m1 = NEG_INF;
        int s = 0;
        for (; s + 2 <= n; s += 2) {
            v2f ka = kk[s], kb = kk[s + 1];
            m0 = fmaxf(m0, fmaf(q0, ka.x, q1 * ka.y));
            m1 = fmaxf(m1, fmaf(q0, kb.x, q1 * kb.y));
        }
        if (s < n) {
            v2f ka = kk[s];
            m0 = fmaxf(m0, fmaf(q0, ka.x, q1 * ka.y));
        }
        const float m = fmaxf(m0, m1);

        // ---- pass 2: exp + accumulate (independent FMA chains) ----
        float l0 = 0.f, l1 = 0.f;
        float x0 = 0.f, x1 = 0.f, z0 = 0.f, z1 = 0.f;
        s = 0;
        for (; s + 2 <= n; s += 2) {
            v2f ka = kk[s], kb = kk[s + 1];
            v2f va = vv[s], vb = vv[s + 1];
            float p0 = __expf(fmaf(q0, ka.x, q1 * ka.y) - m);
            float p1 = __expf(fmaf(q0, kb.x, q1 * kb.y) - m);
            l0 += p0;                l1 += p1;
            x0 = fmaf(p0, va.x, x0); x1 = fmaf(p1, vb.x, x1);
            z0 = fmaf(p0, va.y, z0); z1 = fmaf(p1, vb.y, z1);
        }
        if (s < n) {
            v2f ka = kk[s];
            v2f va = vv[s];
            float p0 = __expf(fmaf(q0, ka.x, q1 * ka.y) - m);
            l0 += p0;
            x0 = fmaf(p0, va.x, x0);
            z0 = fmaf(p0, va.y, z0);
        }

        const float inv = 1.0f / (l0 + l1);
        float* yp = y + ((size_t)b * TT + t) * CC + 2 * h;
        yp[0] = (x0 + x1) * inv;
        yp[1] = (z0 + z1) * inv;
    }
}

// ---------------------------------------------------------------------------
extern "C" void kernel_launch(void* const* d_in, const int* in_sizes, int n_in,
                              void* d_out, int out_size, void* d_ws, size_t ws_size,
                              hipStream_t stream)
{
    (void)in_sizes; (void)n_in; (void)out_size; (void)ws_size;

    const float* x      = (const float*)d_in[0];   // (B,T,C)
    const float* w_qkv  = (const float*)d_in[1];   // (3C,C)
    const float* w_proj = (const float*)d_in[2];   // (C,C)
    const float* qn_w   = (const float*)d_in[3];   // (2,)
    const float* kn_w   = (const float*)d_in[4];   // (2,)
    float* out = (float*)d_out;                    // (B,T,C)

    float* qkv = (float*)d_ws;                     // (B*T, 3C) = 6 MB
    float* y   = qkv + (size_t)BB * TT * QKV_N;    // (B*T, C)  = 2 MB

    const int M = BB * TT;                         // 2048

    // 1) qkv = x @ w_qkv^T : (2048x256)@(256x768)
    {
        int waves = (M / 16) * (QKV_N / 32);       // 3072 waves, 16x32 each
        wmma_gemm_xwt<<<waves / 8, 256, 0, stream>>>(x, w_qkv, qkv, M, QKV_N, CC);
    }

    // 2) RMSNorm q and k head pairs in place
    {
        int n = M * HH;                            // 262144 pairs
        rmsnorm_qk<<<n / 256, 256, 0, stream>>>(qkv, qn_w, kn_w);
    }

    // 3) causal flash attention per (b,h)
    attn_head<<<BB * HH, 256, 0, stream>>>(qkv, y);

    // 4) out = y @ w_proj^T : (2048x256)@(256x256)
    {
        int waves = (M / 16) * (CC / 32);          // 1024 waves
        wmma_gemm_xwt<<<waves / 8, 256, 0, stream>>>(y, w_proj, out, M, CC, CC);
    }
}